// PPTPointNorm_37606733644287
// MI455X (gfx1250) — compile-verified
//
#include <hip/hip_runtime.h>
#include <math.h>

typedef float v2f __attribute__((ext_vector_type(2)));
typedef float v8f __attribute__((ext_vector_type(8)));
typedef float f4  __attribute__((ext_vector_type(4)));

#define CCH  64      // channels C
#define NB   8       // batch B
#define CTXL 256     // context length

// ---------------------------------------------------------------------------
// Kernel A: h[8][128] = silu(token[8][256]) @ W[128][256]^T + bias[128]
// One block, 8 waves; wave w computes the 16x16 tile covering columns
// [16w, 16w+16). M is padded 8 -> 16 with ZERO ROWS IN LDS so the fragment
// loads are unconditional ds_load_b64 (no exec-mask games in the K loop).
//
// Fragment layouts per CDNA5 ISA 7.12.2 (wave32):
//   A (16x4 f32): lane&15 = M row; lanes>=16 hold K=2,3; VGPR0/1 = K lo/hi
//   B (4x16 f32): lane&15 = N col; lanes>=16 hold K=2,3; VGPR0/1 = K lo/hi
//   D (16x16 f32): VGPR r, lanes 0-15 -> (M=r, N=lane)
// ---------------------------------------------------------------------------
__launch_bounds__(256)
__global__ void modulation_gemm_kernel(const float* __restrict__ token,
                                       const float* __restrict__ W,
                                       const float* __restrict__ bias,
                                       float* __restrict__ h) {
  __shared__ float stoken[16 * CTXL];   // 16 KB: rows 0-7 silu(token), 8-15 zero
  const int tid = threadIdx.x;
  for (int i = tid; i < NB * CTXL; i += 256) {
    float v = token[i];
    stoken[i] = v / (1.0f + __expf(-v));
  }
  for (int i = NB * CTXL + tid; i < 16 * CTXL; i += 256) {
    stoken[i] = 0.0f;                   // zero-pad rows M=8..15
  }
  __syncthreads();

  const int wave = tid >> 5;
  const int lane = tid & 31;
  const int half = lane >> 4;           // selects K pair {0,1} vs {2,3}
  const int l    = lane & 15;
  const int n    = wave * 16 + l;       // output column 0..127

  const float* arow = stoken + l * CTXL;        // row l of padded A
  const float* wrow = W + (size_t)n * CTXL;     // B[k][n] = W[n][k]

  v8f acc = {};
#pragma unroll 4
  for (int k0 = 0; k0 < CTXL; k0 += 4) {
    const int k = k0 + half * 2;                // even -> 8B aligned
    const v2f a  = *(const v2f*)(arow + k);     // ds_load_b64
    const v2f bf = *(const v2f*)(wrow + k);     // global_load_b64
    // 8-arg form: (neg_a, A, neg_b, B, c_mod, C, reuse_a, reuse_b)
    acc = __builtin_amdgcn_wmma_f32_16x16x4_f32(false, a, false, bf,
                                                (short)0, acc, false, false);
  }

  if (half == 0) {                 // rows M=0..7 live in VGPRs 0..7, lanes 0..15
    const float bn = bias[n];
#pragma unroll
    for (int r = 0; r < NB; ++r) {
      h[r * (2 * CCH) + n] = acc[r] + bn;
    }
  }
}

// ---------------------------------------------------------------------------
// Kernel B: memory-bound LayerNorm + per-batch modulation.
// 16 lanes per row, each lane owns 4 channels (one float4). Wave32 covers
// 2 rows = 512 contiguous bytes per b128 load -> perfect coalescing.
// 4-step shfl_xor butterfly inside each 16-lane half for sum / sumsq.
// Non-temporal load/store: streams 512 MB, keep it out of L2's way.
// ---------------------------------------------------------------------------
__launch_bounds__(256)
__global__ void pointnorm_mod_kernel(const float* __restrict__ x,
                                     const float* __restrict__ gamma,
                                     const float* __restrict__ beta,
                                     const long long* __restrict__ coors,
                                     const float* __restrict__ h,
                                     float* __restrict__ out,
                                     int n) {
  __shared__ float sg[CCH];
  __shared__ float sb[CCH];
  __shared__ float sh[NB * 2 * CCH];   // 4 KB: [shift(64) | scale(64)] per batch

  const int tid = threadIdx.x;
  if (tid < CCH) { sg[tid] = gamma[tid]; sb[tid] = beta[tid]; }
  for (int i = tid; i < NB * 2 * CCH; i += 256) sh[i] = h[i];
  __syncthreads();

  const int wave = tid >> 5;
  const int lane = tid & 31;
  const int half = lane >> 4;
  const int l    = lane & 15;
  const long long row = (long long)blockIdx.x * 16 + (wave << 1) + half;
  if (row >= n) return;

  const f4 xv = __builtin_nontemporal_load(
      (const f4*)(x + row * CCH + l * 4));

  float s  = xv.x + xv.y + xv.z + xv.w;
  float sq = xv.x * xv.x + xv.y * xv.y + xv.z * xv.z + xv.w * xv.w;
#pragma unroll
  for (int m = 1; m <= 8; m <<= 1) {     // stays within the 16-lane half
    s  += __shfl_xor(s,  m, 32);
    sq += __shfl_xor(sq, m, 32);
  }
  const float mean = s * (1.0f / (float)CCH);
  const float var  = sq * (1.0f / (float)CCH) - mean * mean;
  const float inv  = rsqrtf(var + 1e-5f);

  const int bidx = (int)coors[row * 4];          // coors[:,0], int64
  const float* __restrict__ shf = sh + bidx * (2 * CCH);
  const float* __restrict__ scl = shf + CCH;
  const int c0 = l * 4;

  f4 o;
  o.x = ((xv.x - mean) * inv * sg[c0 + 0] + sb[c0 + 0]) * (1.0f + scl[c0 + 0]) + shf[c0 + 0];
  o.y = ((xv.y - mean) * inv * sg[c0 + 1] + sb[c0 + 1]) * (1.0f + scl[c0 + 1]) + shf[c0 + 1];
  o.z = ((xv.z - mean) * inv * sg[c0 + 2] + sb[c0 + 2]) * (1.0f + scl[c0 + 2]) + shf[c0 + 2];
  o.w = ((xv.w - mean) * inv * sg[c0 + 3] + sb[c0 + 3]) * (1.0f + scl[c0 + 3]) + shf[c0 + 3];

  __builtin_nontemporal_store(o, (f4*)(out + row * CCH + c0));
}

// ---------------------------------------------------------------------------
extern "C" void kernel_launch(void* const* d_in, const int* in_sizes, int n_in,
                              void* d_out, int out_size, void* d_ws, size_t ws_size,
                              hipStream_t stream) {
  const float*     x     = (const float*)d_in[0];
  const float*     token = (const float*)d_in[1];
  const float*     gamma = (const float*)d_in[2];
  const float*     beta  = (const float*)d_in[3];
  const float*     W     = (const float*)d_in[4];
  const float*     bias  = (const float*)d_in[5];
  const long long* coors = (const long long*)d_in[6];
  float* out = (float*)d_out;
  float* h   = (float*)d_ws;                 // (8,128) fp32 = 4 KB scratch

  const int n = in_sizes[0] / CCH;           // number of rows (1e6)

  modulation_gemm_kernel<<<1, 256, 0, stream>>>(token, W, bias, h);

  const int rows_per_block = 16;             // 256 threads, 16 lanes/row
  const int blocks = (n + rows_per_block - 1) / rows_per_block;
  pointnorm_mod_kernel<<<blocks, 256, 0, stream>>>(x, gamma, beta, coors, h, out, n);
}